// PartialSinkhorn_18262200943530
// MI455X (gfx1250) — compile-verified
//
#include <hip/hip_runtime.h>
#include <hip/hip_bf16.h>

// ---- CDNA5 (gfx1250) vector types ----
typedef float v2f  __attribute__((ext_vector_type(2)));
typedef float f4v  __attribute__((ext_vector_type(4)));
typedef float v8f  __attribute__((ext_vector_type(8)));

#define TAU_INV 10.0f
#define BS   64
#define N    1024
#define N2   (N * N)          // 1M elements per batch
#define N4TOT ((size_t)BS * N2 / 4)  // total float4 chunks = 16M

// -----------------------------------------------------------------------------
// Kernel 1: E = exp(-X/tau) into d_out; also init rv = 1.
// X is read once -> non-temporal load. E is hot for 40 passes -> regular store.
// -----------------------------------------------------------------------------
__global__ void ps_exp_init(const float* __restrict__ X,
                            float* __restrict__ E,
                            float* __restrict__ rv) {
    size_t g      = (size_t)blockIdx.x * blockDim.x + threadIdx.x;
    size_t stride = (size_t)gridDim.x * blockDim.x;
    const f4v* __restrict__ X4 = (const f4v*)X;
    f4v* __restrict__ E4 = (f4v*)E;
    for (size_t idx = g; idx < N4TOT; idx += stride) {
        f4v x = __builtin_nontemporal_load(&X4[idx]);
        f4v e;
        e.x = __expf(-TAU_INV * x.x);
        e.y = __expf(-TAU_INV * x.y);
        e.z = __expf(-TAU_INV * x.z);
        e.w = __expf(-TAU_INV * x.w);
        E4[idx] = e;
    }
    if (g < (size_t)BS * N) rv[g] = 1.0f;   // vb = ones
}

// -----------------------------------------------------------------------------
// Kernel 2 (step A): lu[b,i] = 1 / sum_k E[b,i,k] * rv[b,k]
// WMMA f32 16x16x4: wave owns a 16-row tile. Per 8 k-values:
//   lane m      loads E[row, k0  .. k0+3]  (16B contiguous)
//   lane m+16   loads E[row, k0+4.. k0+7]
// A layout (ISA): VGPR0 = {K0 | K2}, VGPR1 = {K1 | K3} across lane halves, so
// WMMA#0 covers global k {k0,k0+1,k0+4,k0+5}, WMMA#1 covers {k0+2,k0+3,k0+6,k0+7}.
// B is rv broadcast across all 16 columns (same half-lane split), so every
// column of D equals the row-sums; accumulate through C/D in f32.
// grid = (8, 64), block = 256 (8 waves -> 128 rows per block)
// -----------------------------------------------------------------------------
__global__ void ps_stepA_wmma(const float* __restrict__ E,
                              const float* __restrict__ rv,
                              float* __restrict__ lu) {
    __shared__ __align__(16) float srv[N];
    const int b = blockIdx.y;

    // stage rv[b,:] into LDS (coalesced float4)
    {
        const f4v* src = (const f4v*)(rv + (size_t)b * N);
        f4v* dst = (f4v*)srv;
        dst[threadIdx.x] = src[threadIdx.x];   // 256 threads * 16B = 4KB
    }
    __syncthreads();

    const int wave = threadIdx.x >> 5;
    const int lane = threadIdx.x & 31;
    const int half = lane >> 4;                 // 0: lanes 0-15, 1: lanes 16-31
    const int i0   = blockIdx.x * 128 + wave * 16;
    const int row  = i0 + (lane & 15);

    const float* __restrict__ Erow = E + ((size_t)b << 20) + ((size_t)row << 10);

    v8f c = {0.f, 0.f, 0.f, 0.f, 0.f, 0.f, 0.f, 0.f};

    #pragma unroll 4
    for (int k0 = 0; k0 < N; k0 += 8) {
        const int kk = k0 + (half << 2);
        f4v a4 = *(const f4v*)(Erow + kk);      // 16B contiguous per lane
        f4v r4 = *(const f4v*)(srv + kk);       // ds_load_b128
        v2f A0 = {a4.x, a4.y}, B0 = {r4.x, r4.y};
        v2f A1 = {a4.z, a4.w}, B1 = {r4.z, r4.w};
        c = __builtin_amdgcn_wmma_f32_16x16x4_f32(false, A0, false, B0,
                                                  (short)0, c, false, false);
        c = __builtin_amdgcn_wmma_f32_16x16x4_f32(false, A1, false, B1,
                                                  (short)0, c, false, false);
    }

    // D layout: VGPR r -> (M=r, N=lane) for lanes 0-15, (M=8+r, N=lane-16) for 16-31.
    // Every column identical (B broadcast) -> read column from lane 0 / lane 16.
    if ((lane & 15) == 0) {
        float* dst = lu + (size_t)b * N + i0 + half * 8;
        #pragma unroll
        for (int r = 0; r < 8; ++r) dst[r] = 1.0f / c[r];
    }
}

// -----------------------------------------------------------------------------
// Kernel 3 (step B): rv[b,j] = min(1, 1 / sum_i E[b,i,j] * lu[b,i])
// Thread-per-column: loads are fully coalesced, lu is an LDS broadcast.
// grid = (4, 64), block = 256 (block covers 256 columns, full i-loop)
// -----------------------------------------------------------------------------
__global__ void ps_stepB(const float* __restrict__ E,
                         const float* __restrict__ lu,
                         float* __restrict__ rv) {
    __shared__ float slu[N];
    const int b = blockIdx.y;
    for (int t = threadIdx.x; t < N; t += 256)
        slu[t] = lu[(size_t)b * N + t];
    __syncthreads();

    const int j = blockIdx.x * 256 + threadIdx.x;
    const float* __restrict__ Ecol = E + ((size_t)b << 20) + j;

    float acc = 0.0f;
    #pragma unroll 8
    for (int i = 0; i < N; ++i)
        acc = fmaf(Ecol[(size_t)i << 10], slu[i], acc);

    rv[(size_t)b * N + j] = fminf(1.0f / acc, 1.0f);
}

// -----------------------------------------------------------------------------
// Kernel 4: out = lu[i] * E * rv[j]  (in place over d_out)
// Result is write-once -> non-temporal store (don't evict anything useful).
// -----------------------------------------------------------------------------
__global__ void ps_finish(float* __restrict__ E,
                          const float* __restrict__ lu,
                          const float* __restrict__ rv) {
    size_t g      = (size_t)blockIdx.x * blockDim.x + threadIdx.x;
    size_t stride = (size_t)gridDim.x * blockDim.x;
    f4v* E4 = (f4v*)E;
    for (size_t idx = g; idx < N4TOT; idx += stride) {
        const size_t b   = idx >> 18;            // 2^18 float4 per batch
        const int    rem = (int)(idx & 262143);
        const int    i   = rem >> 8;             // row
        const int    c4  = (rem & 255) << 2;     // first of 4 columns
        const float  m   = lu[b * N + i];
        f4v r4 = *(const f4v*)(rv + b * N + c4);
        f4v v  = E4[idx];
        v.x *= m * r4.x;  v.y *= m * r4.y;
        v.z *= m * r4.z;  v.w *= m * r4.w;
        __builtin_nontemporal_store(v, &E4[idx]);
    }
}

// -----------------------------------------------------------------------------
extern "C" void kernel_launch(void* const* d_in, const int* in_sizes, int n_in,
                              void* d_out, int out_size, void* d_ws, size_t ws_size,
                              hipStream_t stream) {
    (void)in_sizes; (void)n_in; (void)out_size; (void)ws_size;
    const float* X = (const float*)d_in[0];
    float* E  = (float*)d_out;                 // E lives in d_out, overwritten at end
    float* lu = (float*)d_ws;                  // 64*1024 f32
    float* rv = lu + (size_t)BS * N;           // 64*1024 f32

    ps_exp_init<<<4096, 256, 0, stream>>>(X, E, rv);

    for (int it = 0; it < 20; ++it) {
        ps_stepA_wmma<<<dim3(8, BS), 256, 0, stream>>>(E, rv, lu);
        ps_stepB    <<<dim3(4, BS), 256, 0, stream>>>(E, lu, rv);
    }

    ps_finish<<<4096, 256, 0, stream>>>(E, lu, rv);
}